// AutodiffChannel_21784074126043
// MI455X (gfx1250) — compile-verified
//
#include <hip/hip_runtime.h>
#include <hip/hip_bf16.h>
#include <math.h>

// ---------------------------------------------------------------------------
// EQ cascade: chunked parallel-scan IIR for MI455X (gfx1250, wave32).
//   B=64, C=1, T=262144.  6 cascaded biquads (12-dim linear state).
//   L=256 samples/chunk, 1024 chunks/batch, 2-level scan (32 groups of 32).
//   Tile staging global->LDS via the Tensor Data Mover (TDM) with native
//   row padding (33-float rows, bank-conflict free).
// ---------------------------------------------------------------------------

#define NB      64
#define NT      262144
#define NBAND   6
#define CHLEN   256          // samples per chunk
#define NCHUNK  1024         // chunks per batch  (CHLEN*NCHUNK == NT)
#define NGRP    32           // groups per batch
#define GSZ     32           // chunks per group
#define TT      32           // time-tile for LDS transpose staging
#define LDSP    33           // padded row stride (bank-conflict free)

#if defined(__gfx1250__)
#define USE_TDM 1
#endif

typedef __attribute__((ext_vector_type(2))) float v2f;
typedef __attribute__((ext_vector_type(8))) float v8f;
typedef unsigned int u32x4 __attribute__((ext_vector_type(4)));
typedef int          i32x8 __attribute__((ext_vector_type(8)));
typedef int          i32x4 __attribute__((ext_vector_type(4)));

__device__ __constant__ float PAR_LO[18] = {
  -24.f,   20.f, 0.1f,  -24.f,  200.f, 0.1f,  -24.f,  200.f, 0.1f,
  -24.f, 2000.f, 0.1f,  -24.f, 4000.f, 0.1f,  -24.f, 4000.f, 0.1f };
__device__ __constant__ float PAR_HI[18] = {
   24.f,  200.f, 10.f,   24.f, 2000.f, 10.f,   24.f, 2000.f, 10.f,
   24.f, 8000.f, 10.f,   24.f,12000.f, 10.f,   24.f,12000.f, 10.f };

// ---------------------------------------------------------------------------
// TDM: load one 32x32 f32 tile (row stride CHLEN elements) from global into
// LDS at byte offset lds_off, padding 1 DWORD after every 32 DWORDs stored
// (=> 33-float LDS rows).  D# packing per CDNA5 ISA ch.8 (groups 0/1; groups
// 2/3 zero => dims 3/4 unused).  Wave-level scalar op, tracked by TENSORcnt.
// ---------------------------------------------------------------------------
#ifdef USE_TDM
__device__ __forceinline__
void tdm_load_tile32x32(const float* g, unsigned lds_off)
{
  const unsigned long long ga = (unsigned long long)(uintptr_t)g;
  u32x4 g0;
  g0.x = 1u;                                        // count=1, user descriptor
  g0.y = lds_off;                                   // lds_addr [63:32]
  g0.z = (unsigned)(ga & 0xffffffffu);              // global_addr [95:64]
  g0.w = (unsigned)((ga >> 32) & 0x01ffffffu)       // global_addr [120:96]
       | (2u << 30);                                // type=2 ("image")
  i32x8 g1;
  g1[0] = (int)((2u << 16)      // data_size = 4B
              | (1u << 20)      // pad_enable
              | (4u << 22));    // pad_interval code 4 = every 32 DWORDs
                                // pad_amount  code 0 = 1 DWORD
  g1[1] = (int)(32u << 16);     // tensor_dim0 = 32
  g1[2] = (int)(32u << 16);     // tensor_dim1 = 32
  g1[3] = (int)(32u << 16);     // tile_dim0   = 32
  g1[4] = (int)32u;             // tile_dim1   = 32, tile_dim2 = 0
  g1[5] = (int)CHLEN;           // tensor_dim0_stride = 256 elements
  g1[6] = 0;
  g1[7] = 0;
  const i32x4 gz = {0, 0, 0, 0};
  asm volatile("tensor_load_to_lds %0, %1, %2, %3"
               :
               : "s"(g0), "s"(g1), "s"(gz), "s"(gz)
               : "memory");
}
#endif

// ---------------------------------------------------------------------------
// Kernel 1: parameter projection Z = P(64x22) * W^T(22x18) via WMMA f32
// 16x16x4.  Operands staged zero-padded in LDS so fragment loads are
// branch-free.  8 waves: wave w -> tile (Mtile = w>>1, Ntile = w&1).
// ---------------------------------------------------------------------------
__global__ __launch_bounds__(256)
void params_wmma_kernel(const float* __restrict__ P,    // 64x22
                        const float* __restrict__ W,    // 18x22
                        const float* __restrict__ bias, // 18
                        float* __restrict__ params)     // out: 64x18 (mapped)
{
  __shared__ float Pp[64 * 24];   // P zero-padded to K=24
  __shared__ float Wt[24 * 32];   // W^T zero-padded to 24x32
  const int tid = threadIdx.x;
  for (int idx = tid; idx < 64 * 24; idx += 256) {
    const int m = idx / 24, k = idx % 24;
    Pp[idx] = (k < 22) ? P[m * 22 + k] : 0.f;
  }
  for (int idx = tid; idx < 24 * 32; idx += 256) {
    const int k = idx / 32, n = idx % 32;
    Wt[idx] = (n < 18 && k < 22) ? W[n * 22 + k] : 0.f;
  }
  __syncthreads();

  const int lane = tid & 31;
  const int wave = tid >> 5;               // 0..7
  const int mt   = wave >> 1;              // M tile 0..3
  const int nt   = wave & 1;               // N tile 0..1
  const bool hi  = lane >= 16;
  const int  l16 = lane & 15;
  const int  am  = mt * 16 + l16;
  const int  bn  = nt * 16 + l16;

  v8f acc = {};
  #pragma unroll
  for (int ks = 0; ks < 6; ++ks) {
    const int kb = ks * 4 + (hi ? 2 : 0);  // K of VGPR0; VGPR1 holds kb+1
    v2f a, b;
    a.x = Pp[am * 24 + kb];
    a.y = Pp[am * 24 + kb + 1];
    b.x = Wt[kb * 32 + bn];
    b.y = Wt[(kb + 1) * 32 + bn];
    acc = __builtin_amdgcn_wmma_f32_16x16x4_f32(false, a, false, b,
                                                (short)0, acc, false, false);
  }
  // C/D layout: VGPR r -> row mt*16 + r + (hi?8:0), col nt*16 + l16
  if (bn < 18) {
    const float lo = PAR_LO[bn], hh = PAR_HI[bn];
    #pragma unroll
    for (int r = 0; r < 8; ++r) {
      const int m = mt * 16 + r + (hi ? 8 : 0);
      const float z = acc[r] + bias[bn];
      const float s = 1.f / (1.f + __expf(-z));
      params[m * 18 + bn] = s * (hh - lo) + lo;
    }
  }
}

// ---------------------------------------------------------------------------
// Kernel 2: biquad coefficients per (batch, band).  band 0 = low_shelf,
// band 5 = high_shelf, else peaking.  Stores {b0,b1,b2,a1,a2}/a0.
// ---------------------------------------------------------------------------
__global__ __launch_bounds__(256)
void coeffs_kernel(const float* __restrict__ params, const int* __restrict__ srp,
                   float* __restrict__ coefs)
{
  const int tid = blockIdx.x * blockDim.x + threadIdx.x;
  if (tid >= NB * NBAND) return;
  const int b = tid / NBAND, k = tid % NBAND;
  const float g = params[b * 18 + 3 * k + 0];
  const float f = params[b * 18 + 3 * k + 1];
  const float q = params[b * 18 + 3 * k + 2];
  const float sr = (float)(*srp);

  const float A  = exp2f(g * 0.08304820237218406f);   // 10^(g/40)
  const float w0 = 6.2831853071795864f * (f / sr);
  const float sn = __sinf(w0), cs = __cosf(w0);
  const float alpha = sn / (2.f * q);
  const float sA = sqrtf(A);

  float b0, b1, b2, a0, a1, a2;
  if (k == 0) {            // low shelf
    const float t = 2.f * sA * alpha;
    b0 = A * ((A + 1.f) - (A - 1.f) * cs + t);
    b1 = 2.f * A * ((A - 1.f) - (A + 1.f) * cs);
    b2 = A * ((A + 1.f) - (A - 1.f) * cs - t);
    a0 = (A + 1.f) + (A - 1.f) * cs + t;
    a1 = -2.f * ((A - 1.f) + (A + 1.f) * cs);
    a2 = (A + 1.f) + (A - 1.f) * cs - t;
  } else if (k == NBAND - 1) { // high shelf
    const float t = 2.f * sA * alpha;
    b0 = A * ((A + 1.f) + (A - 1.f) * cs + t);
    b1 = -2.f * A * ((A - 1.f) + (A + 1.f) * cs);
    b2 = A * ((A + 1.f) + (A - 1.f) * cs - t);
    a0 = (A + 1.f) - (A - 1.f) * cs + t;
    a1 = 2.f * ((A - 1.f) - (A + 1.f) * cs);
    a2 = (A + 1.f) - (A - 1.f) * cs - t;
  } else {                 // peaking
    b0 = 1.f + alpha * A;
    b1 = -2.f * cs;
    b2 = 1.f - alpha * A;
    a0 = 1.f + alpha / A;
    a1 = -2.f * cs;
    a2 = 1.f - alpha / A;
  }
  const float inv = 1.f / a0;
  float* c = coefs + tid * 5;
  c[0] = b0 * inv; c[1] = b1 * inv; c[2] = b2 * inv;
  c[3] = a1 * inv; c[4] = a2 * inv;
}

// ---------------------------------------------------------------------------
// One cascade step (transposed direct-form II), 12-dim linear state.
// ---------------------------------------------------------------------------
__device__ __forceinline__
float eq_step(float v, float s1[NBAND], float s2[NBAND], const float cf[NBAND][5])
{
  #pragma unroll
  for (int k = 0; k < NBAND; ++k) {
    const float y  = __builtin_fmaf(cf[k][0], v, s1[k]);
    float n1 = __builtin_fmaf(cf[k][1], v, s2[k]);
    n1 = __builtin_fmaf(-cf[k][3], y, n1);
    float n2 = cf[k][2] * v;
    n2 = __builtin_fmaf(-cf[k][4], y, n2);
    s1[k] = n1; s2[k] = n2; v = y;
  }
  return v;
}

__device__ __forceinline__
void load_coefs(const float* __restrict__ coefs, int batch, float cf[NBAND][5])
{
  #pragma unroll
  for (int k = 0; k < NBAND; ++k)
    #pragma unroll
    for (int j = 0; j < 5; ++j)
      cf[k][j] = coefs[(batch * NBAND + k) * 5 + j];
}

// Gather one 32x32 tile into this wave's padded LDS region.
__device__ __forceinline__
void stage_tile(const float* __restrict__ gsrc, float* __restrict__ sm, int lane)
{
#ifdef USE_TDM
  (void)lane;
  tdm_load_tile32x32(gsrc, (unsigned)(uintptr_t)sm);
  __builtin_amdgcn_s_wait_tensorcnt(0);
#else
  #pragma unroll 8
  for (int i = 0; i < TT; ++i)
    sm[i * LDSP + lane] = gsrc[i * CHLEN + lane];
#endif
}

// ---------------------------------------------------------------------------
// Phase A: lane-per-chunk zero-state pass; record final 12-state per chunk.
// 256 blocks x 256 thr: block = (batch, quarter); wave owns 32 chunks.
// ---------------------------------------------------------------------------
__global__ __launch_bounds__(256)
void phaseA_kernel(const float* __restrict__ x, const float* __restrict__ coefs,
                   float* __restrict__ dstate)
{
  __shared__ float tile[8][TT * LDSP];
  const int lane  = threadIdx.x & 31;
  const int wave  = threadIdx.x >> 5;
  const int batch = blockIdx.x >> 2;
  const int c0    = (blockIdx.x & 3) * 256 + wave * 32;   // first chunk of wave
  const float* xb = x + (size_t)batch * NT;
  float* sm = tile[wave];

  float cf[NBAND][5];
  load_coefs(coefs, batch, cf);
  float s1[NBAND] = {0}, s2[NBAND] = {0};

  for (int t0 = 0; t0 < CHLEN; t0 += TT) {
    stage_tile(xb + c0 * CHLEN + t0, sm, lane);
    __syncthreads();
    #pragma unroll 4
    for (int t = 0; t < TT; ++t)
      (void)eq_step(sm[lane * LDSP + t], s1, s2, cf);
    __syncthreads();
  }
  float* d = dstate + (size_t)(batch * NCHUNK + c0 + lane) * 12;
  #pragma unroll
  for (int k = 0; k < NBAND; ++k) { d[2 * k] = s1[k]; d[2 * k + 1] = s2[k]; }
}

// ---------------------------------------------------------------------------
// Transition probe: column j of T = A^CHLEN by running CHLEN zero-input steps
// from unit state e_j.  64 batches x 12 columns.
// ---------------------------------------------------------------------------
__global__ __launch_bounds__(256)
void transition_kernel(const float* __restrict__ coefs, float* __restrict__ Tmat)
{
  const int tid = blockIdx.x * blockDim.x + threadIdx.x;
  if (tid >= NB * 12) return;
  const int b = tid / 12, j = tid % 12;
  float cf[NBAND][5];
  load_coefs(coefs, b, cf);
  float s1[NBAND], s2[NBAND];
  #pragma unroll
  for (int k = 0; k < NBAND; ++k) {
    s1[k] = (j == 2 * k)     ? 1.f : 0.f;
    s2[k] = (j == 2 * k + 1) ? 1.f : 0.f;
  }
  for (int t = 0; t < CHLEN; ++t) (void)eq_step(0.f, s1, s2, cf);
  #pragma unroll
  for (int k = 0; k < NBAND; ++k) {
    Tmat[b * 144 + (2 * k) * 12 + j]     = s1[k];
    Tmat[b * 144 + (2 * k + 1) * 12 + j] = s2[k];
  }
}

// 12x12 square per batch: dst = src*src.  64 blocks x 144 threads.
__global__ __launch_bounds__(144)
void matsq_kernel(const float* __restrict__ src, float* __restrict__ dst)
{
  const int b = blockIdx.x, i = threadIdx.x / 12, j = threadIdx.x % 12;
  const float* M = src + b * 144;
  float acc = 0.f;
  #pragma unroll
  for (int k = 0; k < 12; ++k) acc = __builtin_fmaf(M[i * 12 + k], M[k * 12 + j], acc);
  dst[b * 144 + i * 12 + j] = acc;
}

// ---------------------------------------------------------------------------
// Phase B scan over chunk states:  s' = T s + d   (T constant per batch).
// ---------------------------------------------------------------------------
__device__ __forceinline__
void matvec_acc(const float Tm[144], float s[12], const float* __restrict__ d)
{
  float ns[12];
  #pragma unroll
  for (int i = 0; i < 12; ++i) {
    float a = d[i];
    #pragma unroll
    for (int k = 0; k < 12; ++k) a = __builtin_fmaf(Tm[i * 12 + k], s[k], a);
    ns[i] = a;
  }
  #pragma unroll
  for (int i = 0; i < 12; ++i) s[i] = ns[i];
}

// B1: group aggregates  gd[b][g] = sum_j T^(31-j) d[g*32+j]
__global__ __launch_bounds__(256)
void phaseB1_kernel(const float* __restrict__ Tmat, const float* __restrict__ dstate,
                    float* __restrict__ gd)
{
  const int tid = blockIdx.x * blockDim.x + threadIdx.x;
  if (tid >= NB * NGRP) return;
  const int b = tid / NGRP, g = tid % NGRP;
  float Tm[144];
  #pragma unroll
  for (int i = 0; i < 144; ++i) Tm[i] = Tmat[b * 144 + i];
  float s[12] = {0};
  for (int j = 0; j < GSZ; ++j)
    matvec_acc(Tm, s, dstate + (size_t)(b * NCHUNK + g * GSZ + j) * 12);
  #pragma unroll
  for (int i = 0; i < 12; ++i) gd[(b * NGRP + g) * 12 + i] = s[i];
}

// B2: per-batch scan of group aggregates with T32 = T^32.
__global__ __launch_bounds__(64)
void phaseB2_kernel(const float* __restrict__ T32, const float* __restrict__ gd,
                    float* __restrict__ ginit)
{
  const int b = threadIdx.x;
  if (b >= NB) return;
  float Tm[144];
  #pragma unroll
  for (int i = 0; i < 144; ++i) Tm[i] = T32[b * 144 + i];
  float s[12] = {0};
  for (int g = 0; g < NGRP; ++g) {
    #pragma unroll
    for (int i = 0; i < 12; ++i) ginit[(b * NGRP + g) * 12 + i] = s[i];
    matvec_acc(Tm, s, gd + (b * NGRP + g) * 12);
  }
}

// B3: expand group inits to per-chunk inits.
__global__ __launch_bounds__(256)
void phaseB3_kernel(const float* __restrict__ Tmat, const float* __restrict__ ginit,
                    const float* __restrict__ dstate, float* __restrict__ cinit)
{
  const int tid = blockIdx.x * blockDim.x + threadIdx.x;
  if (tid >= NB * NGRP) return;
  const int b = tid / NGRP, g = tid % NGRP;
  float Tm[144];
  #pragma unroll
  for (int i = 0; i < 144; ++i) Tm[i] = Tmat[b * 144 + i];
  float s[12];
  #pragma unroll
  for (int i = 0; i < 12; ++i) s[i] = ginit[(b * NGRP + g) * 12 + i];
  for (int j = 0; j < GSZ; ++j) {
    const size_t c = (size_t)(b * NCHUNK + g * GSZ + j) * 12;
    #pragma unroll
    for (int i = 0; i < 12; ++i) cinit[c + i] = s[i];
    matvec_acc(Tm, s, dstate + c);
  }
}

// ---------------------------------------------------------------------------
// Phase C: re-run chunks with correct init states; TDM-staged loads,
// coalesced LDS-staged stores.
// ---------------------------------------------------------------------------
__global__ __launch_bounds__(256)
void phaseC_kernel(const float* __restrict__ x, const float* __restrict__ coefs,
                   const float* __restrict__ cinit, float* __restrict__ out)
{
  __shared__ float tile[8][TT * LDSP];
  const int lane  = threadIdx.x & 31;
  const int wave  = threadIdx.x >> 5;
  const int batch = blockIdx.x >> 2;
  const int c0    = (blockIdx.x & 3) * 256 + wave * 32;
  const float* xb = x + (size_t)batch * NT;
  float* ob = out + (size_t)batch * NT;
  float* sm = tile[wave];

  float cf[NBAND][5];
  load_coefs(coefs, batch, cf);
  float s1[NBAND], s2[NBAND];
  const float* ci = cinit + (size_t)(batch * NCHUNK + c0 + lane) * 12;
  #pragma unroll
  for (int k = 0; k < NBAND; ++k) { s1[k] = ci[2 * k]; s2[k] = ci[2 * k + 1]; }

  for (int t0 = 0; t0 < CHLEN; t0 += TT) {
    stage_tile(xb + c0 * CHLEN + t0, sm, lane);
    __syncthreads();
    #pragma unroll 4
    for (int t = 0; t < TT; ++t) {
      const float v = sm[lane * LDSP + t];
      sm[lane * LDSP + t] = eq_step(v, s1, s2, cf); // in-place: own slot only
    }
    __syncthreads();
    #pragma unroll 8
    for (int i = 0; i < TT; ++i)                    // coalesced scatter
      ob[(c0 + i) * CHLEN + t0 + lane] = sm[i * LDSP + lane];
    __syncthreads();
  }
}

// ---------------------------------------------------------------------------
// Launch
// ---------------------------------------------------------------------------
extern "C" void kernel_launch(void* const* d_in, const int* in_sizes, int n_in,
                              void* d_out, int out_size, void* d_ws, size_t ws_size,
                              hipStream_t stream) {
  const float* x    = (const float*)d_in[0];
  const float* p    = (const float*)d_in[1];
  const float* W    = (const float*)d_in[2];
  const float* bias = (const float*)d_in[3];
  const int*   srp  = (const int*)d_in[4];
  float* out = (float*)d_out;

  float* wsf    = (float*)d_ws;
  float* params = wsf;                 //  1,152 fl
  float* coefs  = wsf + 2048;          //  1,920 fl
  float* Tmat   = wsf + 4096;          //  9,216 fl
  float* Pa     = wsf + 16384;         //  9,216 fl
  float* Pb     = wsf + 32768;         //  9,216 fl
  float* gd     = wsf + 49152;         // 24,576 fl
  float* ginit  = wsf + 81920;         // 24,576 fl
  float* dstate = wsf + 131072;        // 786,432 fl
  float* cinit  = wsf + 1048576;       // 786,432 fl  (total ~7 MiB)

  params_wmma_kernel<<<1, 256, 0, stream>>>(p, W, bias, params);
  coeffs_kernel<<<2, 256, 0, stream>>>(params, srp, coefs);
  phaseA_kernel<<<NB * 4, 256, 0, stream>>>(x, coefs, dstate);
  transition_kernel<<<3, 256, 0, stream>>>(coefs, Tmat);
  matsq_kernel<<<NB, 144, 0, stream>>>(Tmat, Pa);   // T^2
  matsq_kernel<<<NB, 144, 0, stream>>>(Pa, Pb);     // T^4
  matsq_kernel<<<NB, 144, 0, stream>>>(Pb, Pa);     // T^8
  matsq_kernel<<<NB, 144, 0, stream>>>(Pa, Pb);     // T^16
  matsq_kernel<<<NB, 144, 0, stream>>>(Pb, Pa);     // T^32
  phaseB1_kernel<<<8, 256, 0, stream>>>(Tmat, dstate, gd);
  phaseB2_kernel<<<1, 64, 0, stream>>>(Pa, gd, ginit);
  phaseB3_kernel<<<8, 256, 0, stream>>>(Tmat, ginit, dstate, cinit);
  phaseC_kernel<<<NB * 4, 256, 0, stream>>>(x, coefs, cinit, out);
}